// SparseMixerRouter_2302102471511
// MI455X (gfx1250) — compile-verified
//
#include <hip/hip_runtime.h>
#include <hip/hip_bf16.h>

typedef __attribute__((ext_vector_type(2))) float v2f;
typedef __attribute__((ext_vector_type(8))) float v8f;

#define H 4096            // hidden dim (K)
#define KSPLIT 2          // waves per row-group, splitting K
#define KCHUNK (H / KSPLIT)       // 2048 per wave
#define NEXP 64           // experts (N)
#define ROWS_PER_WAVE 16
#define WAVES_PER_BLOCK 8
#define GROUPS_PER_BLOCK (WAVES_PER_BLOCK / KSPLIT)          // 4 row-groups
#define ROWS_PER_BLOCK (ROWS_PER_WAVE * GROUPS_PER_BLOCK)    // 64 rows

// scores = x[N,H] @ W[E,H]^T, then per-row masked-softmax routing:
//   multiplier[row] = 1 / sum_{j unmasked} exp(s_j - max)
//   selected[row]   = argmax_j s_j
// out layout: [0, nrows) = multiplier, [nrows, 2*nrows) = selected (as float)
__launch_bounds__(256)
__global__ void sparsemixer_router_kernel(const float* __restrict__ x,
                                          const float* __restrict__ W,
                                          float* __restrict__ out,
                                          int nrows)
{
    __shared__ float smem[WAVES_PER_BLOCK][ROWS_PER_WAVE][NEXP + 1]; // +1 pad: no bank conflicts

    const int lane  = threadIdx.x & 31;
    const int wave  = threadIdx.x >> 5;
    const int group = wave >> 1;        // row-group within block (0..3)
    const int khalf = wave & 1;         // which K half this wave accumulates
    const int hlf   = lane >> 4;        // half-wave: selects K pair inside fp32 WMMA frags
    const int l16   = lane & 15;        // row (A) / column (B) within the 16x16 tile

    const int row0  = (blockIdx.x * GROUPS_PER_BLOCK + group) * ROWS_PER_WAVE;
    const int kbase = khalf * KCHUNK;

    // A fragment: 16x4 fp32. lane l16 -> row; VGPR0/1 -> K = 2*hlf, 2*hlf+1.
    const float* aptr  = x + (size_t)(row0 + l16) * H + kbase + 2 * hlf;
    // B fragments: 4x16 fp32 per expert tile e0 in {0,16,32,48}; lane l16 -> column (expert).
    const float* bptr0 = W + (size_t)( 0 + l16) * H + kbase + 2 * hlf;
    const float* bptr1 = W + (size_t)(16 + l16) * H + kbase + 2 * hlf;
    const float* bptr2 = W + (size_t)(32 + l16) * H + kbase + 2 * hlf;
    const float* bptr3 = W + (size_t)(48 + l16) * H + kbase + 2 * hlf;

    v8f acc0 = {}; v8f acc1 = {}; v8f acc2 = {}; v8f acc3 = {};

    #pragma unroll 8
    for (int k = 0; k < KCHUNK; k += 4) {
        // x is streamed exactly once: non-temporal so it doesn't evict W from cache.
        v2f a  = __builtin_nontemporal_load((const v2f*)(aptr + k));
        v2f b0 = *(const v2f*)(bptr0 + k);
        v2f b1 = *(const v2f*)(bptr1 + k);
        v2f b2 = *(const v2f*)(bptr2 + k);
        v2f b3 = *(const v2f*)(bptr3 + k);
        // D = A*B + C ; fp32 WMMA 16x16x4, one A fragment reused across 4 expert tiles.
        acc0 = __builtin_amdgcn_wmma_f32_16x16x4_f32(false, a, false, b0, (short)0, acc0, false, false);
        acc1 = __builtin_amdgcn_wmma_f32_16x16x4_f32(false, a, false, b1, (short)0, acc1, false, false);
        acc2 = __builtin_amdgcn_wmma_f32_16x16x4_f32(false, a, false, b2, (short)0, acc2, false, false);
        acc3 = __builtin_amdgcn_wmma_f32_16x16x4_f32(false, a, false, b3, (short)0, acc3, false, false);
    }

    // C/D layout: VGPR v, half h, lane l16 -> element (M = v + 8*h, N = l16 + 16*tile)
    #pragma unroll
    for (int v = 0; v < 8; ++v) {
        const int m = v + 8 * hlf;
        smem[wave][m][l16 +  0] = acc0[v];
        smem[wave][m][l16 + 16] = acc1[v];
        smem[wave][m][l16 + 32] = acc2[v];
        smem[wave][m][l16 + 48] = acc3[v];
    }
    __syncthreads();

    // Per-row routing reduction: even wave of each K-split pair combines the two
    // partial score tiles; lanes 0..15 each own one row of the group's strip.
    if (khalf == 0 && lane < 16) {
        const float* rowA = smem[wave][l16];      // K half 0 partial scores
        const float* rowB = smem[wave + 1][l16];  // K half 1 partial scores

        float mx = rowA[0] + rowB[0];
        int amax = 0;
        #pragma unroll
        for (int e = 1; e < NEXP; ++e) {
            float s = rowA[e] + rowB[e];
            if (s > mx) { mx = s; amax = e; }     // first-max tie-break, like jnp.argmax
        }

        float sum = 0.0f;
        #pragma unroll
        for (int e = 0; e < NEXP; ++e) {
            float s = rowA[e] + rowB[e];
            float factor = fmaxf(fabsf(s), mx);
            // reference: (mx - s)/factor > 2*0.1 -> masked. Multiplicative form has
            // identical semantics for factor==0 (0 > 0 false) and NaN (false).
            bool masked = (mx - s) > 0.2f * factor;
            if (!masked) sum += __expf(s - mx);
        }

        const int row = row0 + l16;
        out[row]         = 1.0f / sum;        // masked_gates at argmax (mask_for_one == 1 in eval)
        out[nrows + row] = (float)amax;       // selected expert index
    }
}

extern "C" void kernel_launch(void* const* d_in, const int* in_sizes, int n_in,
                              void* d_out, int out_size, void* d_ws, size_t ws_size,
                              hipStream_t stream) {
    const float* x = (const float*)d_in[0];   // [N, 4096]
    const float* W = (const float*)d_in[1];   // [64, 4096]
    float* out = (float*)d_out;               // [2*N] : multiplier then selected

    const int nrows = in_sizes[0] / H;        // 8192
    const int grid  = nrows / ROWS_PER_BLOCK; // 128 blocks of 256 threads (8 waves)

    sparsemixer_router_kernel<<<grid, 256, 0, stream>>>(x, W, out, nrows);
}